// DualQDeformableAttention_39152921870879
// MI455X (gfx1250) — compile-verified
//
#include <hip/hip_runtime.h>
#include <hip/hip_bf16.h>

typedef __attribute__((ext_vector_type(2))) float v2f;
typedef __attribute__((ext_vector_type(8))) float v8f;

#define B_    4
#define C_    256
#define HEADS 8
#define NPTS  4
#define DHEAD 32
#define HW_   128
#define NSP   (HW_ * HW_)   /* 16384 tokens per batch image */
#define BN    (B_ * NSP)    /* 65536 rows */
#define APAD  4             /* LDS row padding (dwords), keeps rows 8B-aligned */

// ---------------------------------------------------------------------------
// f32 WMMA GEMM with LDS-staged A tile: D[BN x M] = A[BN x K] * W[K x M] (+bias)
//   Block = 256 threads (8 waves). blockIdx.y = rowTile (16 rows of A/D),
//   blockIdx.x*8 + waveId = colTile (16 cols of W/D).
//   All 8 waves share one 16 x K A-tile staged cooperatively in LDS
//   (coalesced along the contiguous axis of A), then each wave runs the
//   K/4 chain of V_WMMA_F32_16X16X4_F32 reading A fragments as ds_load_b64.
//   ACF: A[row,k] = A[b*K*NSP + k*NSP + n]  (reads NCHW tensors directly)
//   DCF: D[row,m] -> D[b*M*NSP + m*NSP + n] (writes NCHW output directly)
// ---------------------------------------------------------------------------
template <int K, bool ACF, bool DCF>
__global__ __launch_bounds__(256) void gemm_wmma_f32(
    const float* __restrict__ A, const float* __restrict__ Wt,
    const float* __restrict__ bias, float* __restrict__ D, int M)
{
    __shared__ float As[16 * (K + APAD)];

    const int tid  = threadIdx.x;
    const int lane = tid & 31;
    const int wid  = tid >> 5;

    const int rowTile = blockIdx.y;
    const int row0    = rowTile << 4;

    // ---- cooperative, coalesced stage of the 16 x K A tile into LDS ----
    if (ACF) {
        // contiguous axis of A is the row (n) axis: consecutive tids -> consecutive rows
        const int    bb    = row0 >> 14;            // row0 / NSP
        const int    n0    = row0 & (NSP - 1);
        const size_t Abase = (size_t)bb * K * NSP + n0;
#pragma unroll
        for (int i = 0; i < (16 * K) / 256; ++i) {
            const int idx = i * 256 + tid;
            const int r   = idx & 15;
            const int k   = idx >> 4;
            As[r * (K + APAD) + k] = A[Abase + (size_t)k * NSP + r];
        }
    } else {
        // row-major A: consecutive tids -> consecutive k of one row
#pragma unroll
        for (int i = 0; i < (16 * K) / 256; ++i) {
            const int idx = i * 256 + tid;
            const int r   = idx / K;
            const int k   = idx - r * K;
            As[r * (K + APAD) + k] = A[(size_t)(row0 + r) * K + k];
        }
    }
    __syncthreads();

    const int colTiles = M >> 4;
    const int colTile  = blockIdx.x * 8 + wid;      // wave-uniform
    if (colTile < colTiles) {                       // uniform branch: EXEC all-1 inside
        const int col0 = colTile << 4;
        const int half = lane >> 4;                 // 0: K pair {k0,k0+1}; 1: {k0+2,k0+3}
        const int sub  = lane & 15;
        const int bcol = col0 + sub;

        __builtin_prefetch(&Wt[bcol], 0, 0);        // global_prefetch_b8 on weight column

        const float* __restrict__ Arow = &As[sub * (K + APAD)];

        v8f acc = {};
#pragma unroll 8
        for (int k0 = 0; k0 < K; k0 += 4) {
            const int ka = k0 + half * 2;
            const float2 ap = *reinterpret_cast<const float2*>(&Arow[ka]);  // ds_load_b64
            v2f a, b;
            a.x = ap.x;
            a.y = ap.y;
            b.x = Wt[(size_t)ka * M + bcol];
            b.y = Wt[(size_t)(ka + 1) * M + bcol];
            acc = __builtin_amdgcn_wmma_f32_16x16x4_f32(
                /*neg_a=*/false, a, /*neg_b=*/false, b,
                /*c_mod=*/(short)0, acc, /*reuse_a=*/false, /*reuse_b=*/false);
        }

        const float bv = bias ? bias[bcol] : 0.0f;
#pragma unroll
        for (int v = 0; v < 8; ++v) {
            const int   r   = row0 + v + half * 8;
            const float val = acc[v] + bv;
            if (DCF) {
                const int bb = r >> 14;             // r / NSP
                const int nn = r & (NSP - 1);
                D[((size_t)bb * M + bcol) * NSP + nn] = val;
            } else {
                D[(size_t)r * M + bcol] = val;
            }
        }
    }
}

// ---------------------------------------------------------------------------
// In-place softmax over 32 logits per row (flattened heads*points), wave/row.
// ---------------------------------------------------------------------------
__global__ __launch_bounds__(256) void softmax32_kernel(float* __restrict__ aw, int rows)
{
    const int lane = threadIdx.x & 31;
    const int wid  = threadIdx.x >> 5;
    const int row  = blockIdx.x * (blockDim.x >> 5) + wid;
    if (row >= rows) return;

    float v = aw[(size_t)row * 32 + lane];
    float m = v;
#pragma unroll
    for (int off = 16; off > 0; off >>= 1) m = fmaxf(m, __shfl_xor(m, off, 32));
    const float e = __expf(v - m);
    float s = e;
#pragma unroll
    for (int off = 16; off > 0; off >>= 1) s += __shfl_xor(s, off, 32);
    aw[(size_t)row * 32 + lane] = e / s;
}

// ---------------------------------------------------------------------------
// Deformable bilinear sampling + attention-weighted sum.
// One wave per (b, n, head); lane = channel within the head (d=32, contiguous
// 128B in V -> coalesced L2 gathers). Writes into concat buffer (BN x 512).
// ---------------------------------------------------------------------------
__global__ __launch_bounds__(256) void deform_sample_kernel(
    const float* __restrict__ V,    // (BN, 256): V[bn*256 + h*32 + d]
    const float* __restrict__ off,  // (BN, 64):  [h*8 + p*2 + {x,y}]
    const float* __restrict__ aw,   // (BN, 32):  [h*4 + p], softmaxed
    float* __restrict__ cat,        // (BN, 512)
    int catOffset)                  // 0 for branch 1, 256 for branch 2
{
    const int lane = threadIdx.x & 31;
    const int wid  = threadIdx.x >> 5;
    const int gid  = blockIdx.x * (blockDim.x >> 5) + wid;   // (bn, h) flat
    if (gid >= BN * HEADS) return;

    const int h  = gid & (HEADS - 1);
    const int bn = gid >> 3;
    const int b  = bn >> 14;        // / NSP
    const int n  = bn & (NSP - 1);
    const int x  = n & (HW_ - 1);
    const int y  = n >> 7;

    const float rx = (x + 0.5f) / (float)HW_;
    const float ry = (y + 0.5f) / (float)HW_;

    const float* vb = V + (size_t)b * NSP * C_ + h * DHEAD + lane;

    float acc = 0.0f;
#pragma unroll
    for (int p = 0; p < NPTS; ++p) {
        const float ox = off[(size_t)bn * 64 + h * 8 + p * 2 + 0];
        const float oy = off[(size_t)bn * 64 + h * 8 + p * 2 + 1];
        // ix = ((2*(ref+off)) * W - 1) * 0.5 = (ref+off)*W - 0.5
        const float ix = (rx + ox) * (float)HW_ - 0.5f;
        const float iy = (ry + oy) * (float)HW_ - 0.5f;
        const float x0f = floorf(ix), y0f = floorf(iy);
        const int   x0 = (int)x0f,   y0 = (int)y0f;
        const float wx1 = ix - x0f,  wy1 = iy - y0f;
        const float wx0 = 1.0f - wx1, wy0 = 1.0f - wy1;

        auto corner = [&](int xi, int yi, float w) -> float {
            if (xi < 0 || xi >= HW_ || yi < 0 || yi >= HW_) return 0.0f;
            return w * vb[(size_t)((yi << 7) + xi) * C_];
        };

        float s = corner(x0,     y0,     wx0 * wy0)
                + corner(x0 + 1, y0,     wx1 * wy0)
                + corner(x0,     y0 + 1, wx0 * wy1)
                + corner(x0 + 1, y0 + 1, wx1 * wy1);

        acc += aw[(size_t)bn * 32 + h * 4 + p] * s;
    }

    cat[(size_t)bn * 512 + catOffset + h * DHEAD + lane] = acc;
}

// ---------------------------------------------------------------------------
// Launch
// ---------------------------------------------------------------------------
extern "C" void kernel_launch(void* const* d_in, const int* in_sizes, int n_in,
                              void* d_out, int out_size, void* d_ws, size_t ws_size,
                              hipStream_t stream)
{
    const float* x1    = (const float*)d_in[0];
    const float* x2    = (const float*)d_in[1];
    const float* x3    = (const float*)d_in[2];
    const float* Wv    = (const float*)d_in[3];
    const float* Woff1 = (const float*)d_in[4];
    const float* boff1 = (const float*)d_in[5];
    const float* Woff2 = (const float*)d_in[6];
    const float* boff2 = (const float*)d_in[7];
    const float* Waw1  = (const float*)d_in[8];
    const float* baw1  = (const float*)d_in[9];
    const float* Waw2  = (const float*)d_in[10];
    const float* baw2  = (const float*)d_in[11];
    const float* Wout  = (const float*)d_in[12];
    const float* bout  = (const float*)d_in[13];
    float* out = (float*)d_out;

    // workspace (floats): V(BN*256) off1(BN*64) off2(BN*64) aw1(BN*32) aw2(BN*32) cat(BN*512)
    float* V    = (float*)d_ws;
    float* off1 = V    + (size_t)BN * 256;
    float* off2 = off1 + (size_t)BN * 64;
    float* aw1  = off2 + (size_t)BN * 64;
    float* aw2  = aw1  + (size_t)BN * 32;
    float* cat  = aw2  + (size_t)BN * 32;

    const int THREADS  = 256;
    const int WPB      = THREADS / 32;   // 8 waves per block
    const int rowTiles = BN >> 4;        // 4096
    auto grid = [&](int M) { return dim3(((M >> 4) + 7) / 8, rowTiles, 1); };

    // V = x3f @ Wv                        (A channel-first, D row-major)
    gemm_wmma_f32<256, true,  false><<<grid(256), THREADS, 0, stream>>>(x3, Wv, nullptr, V, 256);
    // offsets / attention logits (branch 1 and 2)
    gemm_wmma_f32<256, true,  false><<<grid(64),  THREADS, 0, stream>>>(x1, Woff1, boff1, off1, 64);
    gemm_wmma_f32<256, true,  false><<<grid(32),  THREADS, 0, stream>>>(x1, Waw1,  baw1,  aw1,  32);
    gemm_wmma_f32<256, true,  false><<<grid(64),  THREADS, 0, stream>>>(x2, Woff2, boff2, off2, 64);
    gemm_wmma_f32<256, true,  false><<<grid(32),  THREADS, 0, stream>>>(x2, Waw2,  baw2,  aw2,  32);

    // softmax over the 32 flattened (head,point) logits
    softmax32_kernel<<<(BN + WPB - 1) / WPB, THREADS, 0, stream>>>(aw1, BN);
    softmax32_kernel<<<(BN + WPB - 1) / WPB, THREADS, 0, stream>>>(aw2, BN);

    // deformable sampling + weighted sum, one wave per (b,n,head)
    const int sampBlocks = (BN * HEADS + WPB - 1) / WPB;
    deform_sample_kernel<<<sampBlocks, THREADS, 0, stream>>>(V, off1, aw1, cat, 0);
    deform_sample_kernel<<<sampBlocks, THREADS, 0, stream>>>(V, off2, aw2, cat, 256);

    // out = cat @ Wout + bout             (A row-major, D channel-first -> NCHW)
    gemm_wmma_f32<512, false, true><<<grid(256), THREADS, 0, stream>>>(cat, Wout, bout, out, 256);
}